// ImageTreensformerV5_1898375545717
// MI455X (gfx1250) — compile-verified
//
#include <hip/hip_runtime.h>

typedef __attribute__((ext_vector_type(2))) float v2f;
typedef __attribute__((ext_vector_type(8))) float v8f;

#define NLEV  7
#define NB    32      // batch
#define NC    3       // channels
#define HW    4096    // 64*64
#define NEMB  896
#define INNER 128     // 896/7
#define NCLS  1000

// ---------------------------------------------------------------------------
// Kernel 1: m[b*3+c] = mean over H*W of x[b,c,:,:]   (x is contiguous BCHW)
// ---------------------------------------------------------------------------
__global__ void itf_means_kernel(const float* __restrict__ x,
                                 float* __restrict__ m) {
    __shared__ float red[256];
    const int bc = blockIdx.x;                // 0..95
    const float* p = x + (size_t)bc * HW;
    float s = 0.0f;
    for (int i = threadIdx.x; i < HW; i += 256) s += p[i];
    red[threadIdx.x] = s;
    __syncthreads();
    for (int off = 128; off > 0; off >>= 1) {
        if ((int)threadIdx.x < off) red[threadIdx.x] += red[threadIdx.x + off];
        __syncthreads();
    }
    if (threadIdx.x == 0) m[bc] = red[0] * (1.0f / (float)HW);
}

// ---------------------------------------------------------------------------
// Kernel 2: root_avg[b][j] = sum_c m[b][c] * (sum_l emb_w[(l*3+c)*896 + 768+j])
//                            + emb_b[768+j]
// (exactly the spatial mean of the level-6 slice of emb, by linearity)
// ---------------------------------------------------------------------------
__global__ void itf_rootavg_kernel(const float* __restrict__ m,
                                   const float* __restrict__ emb_w,
                                   const float* __restrict__ emb_b,
                                   float* __restrict__ ra) {
    const int idx = blockIdx.x * blockDim.x + threadIdx.x;   // 0..4095
    const int b = idx >> 7;         // / 128
    const int j = idx & 127;
    float acc = emb_b[(NLEV - 1) * INNER + j];
    #pragma unroll
    for (int c = 0; c < NC; ++c) {
        float w2 = 0.0f;
        #pragma unroll
        for (int l = 0; l < NLEV; ++l)
            w2 += emb_w[(size_t)(l * NC + c) * NEMB + (NLEV - 1) * INNER + j];
        acc = fmaf(m[b * NC + c], w2, acc);
    }
    ra[idx] = acc;
}

// ---------------------------------------------------------------------------
// Kernel 3: out(32x1000) = ra(32x128) @ cls_w(128x1000) + cls_b
// One wave32 per 16x16 output tile; K swept in steps of 4 with
// V_WMMA_F32_16X16X4_F32 (full fp32 matrix pipe, exact semantics).
// ---------------------------------------------------------------------------
__global__ void itf_gemm_wmma_kernel(const float* __restrict__ ra,
                                     const float* __restrict__ cls_w,
                                     const float* __restrict__ cls_b,
                                     float* __restrict__ out) {
    const int NT = (NCLS + 15) / 16;                  // 63 N-tiles
    const int wave = (blockIdx.x * blockDim.x + threadIdx.x) >> 5;
    const int lane = threadIdx.x & 31;
    if (wave >= 2 * NT) return;                       // whole wave exits together
    const int mt = wave / NT;                         // 0..1
    const int nt = wave % NT;                         // 0..62

    const int lrow = lane & 15;    // M (for A) / N (for B,C/D) within tile
    const int hi   = lane >> 4;    // lane-group select (K pairs / M halves)

    // Branchless N-edge handling: clamp column, zero contribution.
    const int   n     = nt * 16 + lrow;
    const int   ncl   = (n < NCLS) ? n : (NCLS - 1);
    const float nmask = (n < NCLS) ? 1.0f : 0.0f;

    const float* arow = ra + (size_t)(mt * 16 + lrow) * INNER;

    v8f acc = {};
    for (int k0 = 0; k0 < INNER; k0 += 4) {
        const int ka = k0 + hi * 2;
        v2f a, b;
        a.x = arow[ka + 0];                                   // A[M=lrow][ka]
        a.y = arow[ka + 1];                                   // A[M=lrow][ka+1]
        b.x = nmask * cls_w[(size_t)(ka + 0) * NCLS + ncl];   // B[ka  ][N=n]
        b.y = nmask * cls_w[(size_t)(ka + 1) * NCLS + ncl];   // B[ka+1][N=n]
        // 8 args: (neg_a, A, neg_b, B, c_mod, C, reuse_a, reuse_b)
        acc = __builtin_amdgcn_wmma_f32_16x16x4_f32(
                  false, a, false, b, (short)0, acc, false, false);
    }

    const float bias = nmask * cls_b[ncl];
    #pragma unroll
    for (int r = 0; r < 8; ++r) {
        const int row = mt * 16 + hi * 8 + r;                 // C/D layout
        if (n < NCLS) out[(size_t)row * NCLS + n] = acc[r] + bias;
    }
}

// ---------------------------------------------------------------------------
// Launch
// ---------------------------------------------------------------------------
extern "C" void kernel_launch(void* const* d_in, const int* in_sizes, int n_in,
                              void* d_out, int out_size, void* d_ws, size_t ws_size,
                              hipStream_t stream) {
    const float* x     = (const float*)d_in[0];   // (32,3,64,64)
    const float* emb_w = (const float*)d_in[1];   // (21,896)
    const float* emb_b = (const float*)d_in[2];   // (896,)
    const float* cls_w = (const float*)d_in[3];   // (128,1000)
    const float* cls_b = (const float*)d_in[4];   // (1000,)
    float* out = (float*)d_out;                   // (32,1000)

    float* ws = (float*)d_ws;
    float* m  = ws;            // 96 floats
    float* ra = ws + 128;      // 32*128 floats (aligned slot)

    itf_means_kernel<<<NB * NC, 256, 0, stream>>>(x, m);
    itf_rootavg_kernel<<<(NB * INNER) / 256, 256, 0, stream>>>(m, emb_w, emb_b, ra);

    const int ntiles = 2 * ((NCLS + 15) / 16);    // 126 waves
    const int threads = 128;                      // 4 waves/block
    const int blocks = (ntiles * 32 + threads - 1) / threads;
    itf_gemm_wmma_kernel<<<blocks, threads, 0, stream>>>(ra, cls_w, cls_b, out);
}